// BiRNNLM_40630390620358
// MI455X (gfx1250) — compile-verified
//
#include <hip/hip_runtime.h>

typedef __attribute__((ext_vector_type(2))) float v2f;
typedef __attribute__((ext_vector_type(8))) float v8f;

#define SEQ   64
#define BATCH 32
#define VOCAB 32000
#define EMBED 32
#define HID   16
// ws layout (floats): [0..63] per-t softmax sums, [64..64+64*32*16) hidden states H

// ---------------------------------------------------------------------------
// Kernel A: sequential RNN recurrence. One block of 256 threads:
// thread = (b, jg): b = batch row (32), jg picks 2 of the 16 hidden outputs.
// Serial chain: 64 steps x ~96 FMA per thread. Also zeros sums[64].
// ---------------------------------------------------------------------------
__global__ void rnn_recur(const int* __restrict__ tok,
                          const float* __restrict__ h0,
                          const float* __restrict__ we,
                          const float* __restrict__ i2h,
                          float* __restrict__ ws) {
  __shared__ float sW[(EMBED + HID) * HID];   // 768 floats: i2h
  __shared__ float sH[BATCH * HID];           // current hidden state
  const int tid = threadIdx.x;

  for (int i = tid; i < (EMBED + HID) * HID; i += blockDim.x) sW[i] = i2h[i];
  if (tid < BATCH * HID) sH[tid] = h0[tid];
  if (tid < SEQ) ws[tid] = 0.0f;              // zero softmax sums every launch
  __syncthreads();

  const int b  = tid >> 3;        // 0..31
  const int jg = tid & 7;         // 0..7
  const int j0 = jg * 2;
  const int j1 = j0 + 1;
  float* Hst = ws + 64;

  for (int t = 0; t < SEQ; ++t) {
    const int token = tok[t * BATCH + b];
    const float* er = we + (long)token * EMBED;
    float acc0 = 0.f, acc1 = 0.f;
    #pragma unroll
    for (int k = 0; k < EMBED; ++k) {
      const float e = er[k];
      acc0 += e * sW[k * HID + j0];
      acc1 += e * sW[k * HID + j1];
    }
    #pragma unroll
    for (int k = 0; k < HID; ++k) {
      const float hv = sH[b * HID + k];
      acc0 += hv * sW[(EMBED + k) * HID + j0];
      acc1 += hv * sW[(EMBED + k) * HID + j1];
    }
    const float h0n = tanhf(acc0);
    const float h1n = tanhf(acc1);
    __syncthreads();                        // all reads of sH done
    sH[b * HID + j0] = h0n;
    sH[b * HID + j1] = h1n;
    Hst[t * BATCH * HID + b * HID + j0] = h0n;
    Hst[t * BATCH * HID + b * HID + j1] = h1n;
    __syncthreads();                        // writes visible before next step
  }
}

// ---------------------------------------------------------------------------
// WMMA slab: one wave computes a 32x16 slab of logits[t] = H[t] @ i2o for
// one 16-column chunk, K=16 via 4 chained V_WMMA_F32_16X16X4_F32 per M-tile.
// A 16x4 layout:  lane L: M = L%16; VGPR0 = K{0 | 2}, VGPR1 = K{1 | 3} by half.
// B 4x16 layout:  lane L: N = L%16; same K striping.
// C/D 16x16:      VGPR i: M = i + 8*(L>=16), N = L%16.
// ---------------------------------------------------------------------------
__device__ __forceinline__ void slab_wmma(const float* __restrict__ Ht,
                                          const float* __restrict__ i2o,
                                          int col, int lane,
                                          v8f& c0, v8f& c1) {
  const int half = lane >> 4;       // 0 or 1
  const int m    = lane & 15;
  v8f z = {0.f, 0.f, 0.f, 0.f, 0.f, 0.f, 0.f, 0.f};
  c0 = z; c1 = z;
  #pragma unroll
  for (int kk = 0; kk < HID; kk += 4) {
    const int k0 = kk + half * 2;
    v2f a0, a1, bb;
    a0.x = Ht[m * HID + k0];
    a0.y = Ht[m * HID + k0 + 1];
    a1.x = Ht[(m + 16) * HID + k0];
    a1.y = Ht[(m + 16) * HID + k0 + 1];
    bb.x = i2o[(size_t)k0 * VOCAB + col];
    bb.y = i2o[(size_t)(k0 + 1) * VOCAB + col];
    c0 = __builtin_amdgcn_wmma_f32_16x16x4_f32(false, a0, false, bb,
                                               (short)0, c0, false, false);
    c1 = __builtin_amdgcn_wmma_f32_16x16x4_f32(false, a1, false, bb,
                                               (short)0, c1, false, false);
  }
}

// ---------------------------------------------------------------------------
// Kernel B: per-timestep global sum of exp(logits). No big stores: i2o stays
// L2-resident, only one float atomic per block reaches memory.
// grid = (250, 64), block = 256 (8 waves; wave w covers columns of chunk
// blockIdx.x*8+w), t = blockIdx.y.
// ---------------------------------------------------------------------------
__global__ void sum_pass(const float* __restrict__ ws_h,   // ws base
                         const float* __restrict__ i2o,
                         float* __restrict__ sums) {       // == ws base
  const int t    = blockIdx.y;
  const int wave = threadIdx.x >> 5;
  const int lane = threadIdx.x & 31;
  const int col  = (blockIdx.x * 8 + wave) * 16 + (lane & 15);
  const float* Ht = ws_h + 64 + t * BATCH * HID;

  v8f c0, c1;
  slab_wmma(Ht, i2o, col, lane, c0, c1);

  float local = 0.f;
  #pragma unroll
  for (int i = 0; i < 8; ++i) local += __expf(c0[i]) + __expf(c1[i]);

  __shared__ float red[256];
  red[threadIdx.x] = local;
  __syncthreads();
  #pragma unroll
  for (int s = 128; s > 0; s >>= 1) {
    if (threadIdx.x < s) red[threadIdx.x] += red[threadIdx.x + s];
    __syncthreads();
  }
  if (threadIdx.x == 0) atomicAdd(&sums[t], red[0]);
}

// ---------------------------------------------------------------------------
// Kernel C: recompute logits (cheap, L2-fed), write normalized exp once.
// Output stores are non-temporal: the 262 MB write-once stream must not
// evict the hot 2 MB i2o matrix from the 192 MB L2.
// HBM traffic ~= output size (262 MB) -> ~11 us floor at 23.3 TB/s.
// ---------------------------------------------------------------------------
__global__ void write_pass(const float* __restrict__ ws,
                           const float* __restrict__ i2o,
                           float* __restrict__ out) {
  const int t    = blockIdx.y;
  const int wave = threadIdx.x >> 5;
  const int lane = threadIdx.x & 31;
  const int half = lane >> 4;
  const int col  = (blockIdx.x * 8 + wave) * 16 + (lane & 15);
  const float* Ht = ws + 64 + t * BATCH * HID;

  v8f c0, c1;
  slab_wmma(Ht, i2o, col, lane, c0, c1);

  const float scale = 1.0f / ws[t];
  float* outT = out + (size_t)t * BATCH * VOCAB;
  #pragma unroll
  for (int i = 0; i < 8; ++i) {
    const int b0 = i + half * 8;                 // rows 0..15 (tile 0)
    __builtin_nontemporal_store(__expf(c0[i]) * scale,
                                &outT[(size_t)b0 * VOCAB + col]);
    __builtin_nontemporal_store(__expf(c1[i]) * scale,
                                &outT[(size_t)(b0 + 16) * VOCAB + col]);
  }
}

extern "C" void kernel_launch(void* const* d_in, const int* in_sizes, int n_in,
                              void* d_out, int out_size, void* d_ws, size_t ws_size,
                              hipStream_t stream) {
  const int*   tok = (const int*)  d_in[0];   // [64,32]
  const float* h0  = (const float*)d_in[1];   // [32,16]
  const float* we  = (const float*)d_in[2];   // [32000,32]
  const float* i2h = (const float*)d_in[3];   // [48,16]
  const float* i2o = (const float*)d_in[4];   // [16,32000]
  float*       out = (float*)d_out;           // [64,32,32000]
  float*       ws  = (float*)d_ws;            // sums[64] + H[64*32*16]

  rnn_recur<<<1, 256, 0, stream>>>(tok, h0, we, i2h, ws);

  dim3 grid(VOCAB / (8 * 16), SEQ);           // (250, 64)
  sum_pass<<<grid, 256, 0, stream>>>(ws, i2o, ws);
  write_pass<<<grid, 256, 0, stream>>>(ws, i2o, out);
}